// OpenSetGenerater_22625887715979
// MI455X (gfx1250) — compile-verified
//
#include <hip/hip_runtime.h>
#include <math.h>

// Problem dims (from reference)
#define BS    32
#define NWAY  5
#define NBASE 351
#define FD    640
#define SD    300
#define BROWS (BS * NWAY)   // 160
#define LDATT 352           // padded leading dim for attn rows (8B-aligned vector loads)

typedef __attribute__((ext_vector_type(2))) float v2f;
typedef __attribute__((ext_vector_type(8))) float v8f;

enum { EPI_NONE = 0, EPI_LEAKY = 1, EPI_SIGP1 = 2, EPI_GATE = 3, EPI_RES = 4 };

__device__ __forceinline__ v8f wmma4(v2f a, v2f b, v8f c) {
    return __builtin_amdgcn_wmma_f32_16x16x4_f32(
        /*neg_a=*/false, a, /*neg_b=*/false, b,
        /*c_mod=*/(short)0, c, /*reuse_a=*/false, /*reuse_b=*/false);
}

// Accumulate a 16x64 strip (4 N-tiles) of C += A @ op(B) with f32 WMMA.
// A row-major (M,K), row index pre-clamped (mc). B: transB=0 -> W (N,K) row-major
// (y = x @ W^T); transB=1 -> (K,N) row-major. nc[] are pre-clamped column indices.
// Branch-free: no EXEC-divergent loads; K%4 handled by a masked tail WMMA.
__device__ __forceinline__ void gemm_strip_accum(
    v8f c[4], const float* __restrict__ A, int lda,
    const float* __restrict__ B, int ldb, int transB, int K,
    int mc, const int nc[4], int lhalf)
{
    const float* Arow = A + (size_t)mc * lda + lhalf * 2;   // even offset: 8B aligned
    if (transB) {
        const float* Bc0 = B + nc[0] + (size_t)(lhalf * 2) * ldb;
        const float* Bc1 = B + nc[1] + (size_t)(lhalf * 2) * ldb;
        const float* Bc2 = B + nc[2] + (size_t)(lhalf * 2) * ldb;
        const float* Bc3 = B + nc[3] + (size_t)(lhalf * 2) * ldb;
        for (int k = 0; k + 4 <= K; k += 4) {
            const v2f a = *(const v2f*)(Arow + k);
            const size_t ko = (size_t)k * ldb;
            v2f b0, b1, b2, b3;
            b0.x = Bc0[ko]; b0.y = Bc0[ko + ldb];
            b1.x = Bc1[ko]; b1.y = Bc1[ko + ldb];
            b2.x = Bc2[ko]; b2.y = Bc2[ko + ldb];
            b3.x = Bc3[ko]; b3.y = Bc3[ko + ldb];
            c[0] = wmma4(a, b0, c[0]);
            c[1] = wmma4(a, b1, c[1]);
            c[2] = wmma4(a, b2, c[2]);
            c[3] = wmma4(a, b3, c[3]);
        }
    } else {
        const float* Bc0 = B + (size_t)nc[0] * ldb + lhalf * 2;
        const float* Bc1 = B + (size_t)nc[1] * ldb + lhalf * 2;
        const float* Bc2 = B + (size_t)nc[2] * ldb + lhalf * 2;
        const float* Bc3 = B + (size_t)nc[3] * ldb + lhalf * 2;
        for (int k = 0; k + 4 <= K; k += 4) {
            const v2f a = *(const v2f*)(Arow + k);
            const v2f b0 = *(const v2f*)(Bc0 + k);
            const v2f b1 = *(const v2f*)(Bc1 + k);
            const v2f b2 = *(const v2f*)(Bc2 + k);
            const v2f b3 = *(const v2f*)(Bc3 + k);
            c[0] = wmma4(a, b0, c[0]);
            c[1] = wmma4(a, b1, c[1]);
            c[2] = wmma4(a, b2, c[2]);
            c[3] = wmma4(a, b3, c[3]);
        }
    }
    // Tail for K % 4 != 0: clamp K-indices, zero-weight the A fragment beyond K
    // (B garbage * 0 contributes nothing). Executed at most once, branch-free loads.
    if (K & 3) {
        const int k  = K & ~3;
        const int kk = k + lhalf * 2;
        const int k0 = kk     < K ? kk     : K - 1;
        const int k1 = kk + 1 < K ? kk + 1 : K - 1;
        const float w0 = kk     < K ? 1.0f : 0.0f;
        const float w1 = kk + 1 < K ? 1.0f : 0.0f;
        v2f a;
        a.x = A[(size_t)mc * lda + k0] * w0;
        a.y = A[(size_t)mc * lda + k1] * w1;
#pragma unroll
        for (int j = 0; j < 4; ++j) {
            v2f b;
            if (transB) { b.x = B[(size_t)k0 * ldb + nc[j]]; b.y = B[(size_t)k1 * ldb + nc[j]]; }
            else        { b.x = B[(size_t)nc[j] * ldb + k0]; b.y = B[(size_t)nc[j] * ldb + k1]; }
            c[j] = wmma4(a, b, c[j]);
        }
    }
}

// Generic f32 WMMA GEMM: D[z] = epi( A[z] @ op(B[z]) + A2[z] @ op(B2[z]) + bias ).
// One 16x64 output strip per wave; 4 waves (128 threads) per block.
__global__ void wmma_gemm(
    const float* __restrict__ A,  int lda,  long zsA,
    const float* __restrict__ B,  int ldb,  long zsB,  int transB,  int K,
    const float* __restrict__ A2, int lda2, long zsA2,
    const float* __restrict__ B2, int ldb2, long zsB2, int transB2, int K2,
    const float* __restrict__ bias,
    const float* __restrict__ res,  int ldres, long zsRes,
    const float* __restrict__ gmul, int ldg,   int gdiv, float scale,
    float* __restrict__ D, int ldd, long zsD,
    int M, int N, int epi)
{
    const int lane  = threadIdx.x & 31;
    const int wv    = threadIdx.x >> 5;
    const int strip = blockIdx.x * 4 + wv;     // 64 output columns per wave
    if (strip * 64 >= N) return;               // wave-uniform early-out
    const int tm    = blockIdx.y;
    const int z     = blockIdx.z;
    const int lhalf = lane >> 4;               // 0 = lanes 0-15, 1 = lanes 16-31
    const int li    = lane & 15;

    const int m  = tm * 16 + li;
    const int mc = m < M ? m : M - 1;          // clamp: OOB rows feed unstored outputs
    int nc[4];
#pragma unroll
    for (int j = 0; j < 4; ++j) {
        const int n = strip * 64 + j * 16 + li;
        nc[j] = n < N ? n : N - 1;
    }

    v8f c[4];
#pragma unroll
    for (int j = 0; j < 4; ++j) c[j] = (v8f){0.f, 0.f, 0.f, 0.f, 0.f, 0.f, 0.f, 0.f};

    gemm_strip_accum(c, A + (size_t)z * zsA, lda, B + (size_t)z * zsB, ldb,
                     transB, K, mc, nc, lhalf);
    if (A2)
        gemm_strip_accum(c, A2 + (size_t)z * zsA2, lda2, B2 + (size_t)z * zsB2, ldb2,
                         transB2, K2, mc, nc, lhalf);

    const int dmBase = tm * 16 + lhalf * 8;    // C/D layout per ISA 7.12.2
#pragma unroll
    for (int j = 0; j < 4; ++j) {
        const int dn  = strip * 64 + j * 16 + li;
        const int dnc = nc[j];
        const float bv = bias ? bias[dnc] : 0.0f;
#pragma unroll
        for (int i = 0; i < 8; ++i) {
            const int dm  = dmBase + i;
            const int dmc = dm < M ? dm : M - 1;
            const bool ok = (dm < M) && (dn < N);
            const float v = c[j][i];
            float o;
            if (epi == EPI_LEAKY)      { const float t = v + bv; o = t > 0.0f ? t : 0.1f * t; }
            else if (epi == EPI_SIGP1) { const float t = v + bv; o = 1.0f + 1.0f / (1.0f + __expf(-t)); }
            else if (epi == EPI_GATE)  { o = v * gmul[(size_t)(dmc / gdiv) * ldg + dnc] * scale; }
            else if (epi == EPI_RES)   { o = v + res[(size_t)z * zsRes + (size_t)dmc * ldres + dnc]; }
            else                       { o = v + bv; }
            if (ok) D[(size_t)z * zsD + (size_t)dm * ldd + dn] = o;
        }
    }
}

// avg[b][0:640] = mean_j base_weights[b][j][:]; avg[b][640:940] = mean_j base_seman[b][j][:]
__global__ void avg_kernel(const float* __restrict__ bw,
                           const float* __restrict__ bsem,
                           float* __restrict__ avg)
{
    const int b = blockIdx.x;
    for (int n = threadIdx.x; n < FD + SD; n += blockDim.x) {
        float s = 0.0f;
        if (n < FD) {
            const float* p = bw + (size_t)b * NBASE * FD + n;
            for (int j = 0; j < NBASE; ++j) s += p[(size_t)j * FD];
        } else {
            const float* p = bsem + (size_t)b * NBASE * SD + (n - FD);
            for (int j = 0; j < NBASE; ++j) s += p[(size_t)j * SD];
        }
        avg[(size_t)b * (FD + SD) + n] = s * (1.0f / (float)NBASE);
    }
}

// In-place softmax over rows of length NBASE, leading dim LDATT (one block per row).
__global__ void softmax_kernel(float* __restrict__ s)
{
    __shared__ float red[256];
    float* row = s + (size_t)blockIdx.x * LDATT;
    const int t = threadIdx.x;
    float mx = -3.0e38f;
    for (int j = t; j < NBASE; j += 256) mx = fmaxf(mx, row[j]);
    red[t] = mx; __syncthreads();
    for (int o = 128; o > 0; o >>= 1) { if (t < o) red[t] = fmaxf(red[t], red[t + o]); __syncthreads(); }
    mx = red[0]; __syncthreads();
    float sum = 0.0f;
    for (int j = t; j < NBASE; j += 256) { const float e = __expf(row[j] - mx); row[j] = e; sum += e; }
    red[t] = sum; __syncthreads();
    for (int o = 128; o > 0; o >>= 1) { if (t < o) red[t] += red[t + o]; __syncthreads(); }
    const float inv = 1.0f / red[0];
    __syncthreads();
    for (int j = t; j < NBASE; j += 256) row[j] *= inv;
}

// fake = mean over NWAY of out rows; fake_open likewise; copy support_center.
__global__ void finalize_kernel(const float* __restrict__ outv,
                                const float* __restrict__ outo,
                                const float* __restrict__ sc,
                                float* __restrict__ d_out)
{
    const int tid = blockIdx.x * blockDim.x + threadIdx.x;
    if (tid < BS * FD) {
        const int b = tid / FD, d = tid % FD;
        float s1 = 0.0f, s2 = 0.0f;
        for (int w = 0; w < NWAY; ++w) {
            const size_t r = (size_t)(b * NWAY + w) * FD + d;
            s1 += outv[r];
            s2 += outo[r];
        }
        d_out[tid]           = s1 * (1.0f / (float)NWAY);
        d_out[BS * FD + tid] = s2 * (1.0f / (float)NWAY);
    }
    if (tid < BS * NWAY * FD) d_out[2 * BS * FD + tid] = sc[tid];
}

static inline void launch_gemm(hipStream_t st,
    const float* A, int lda, long zsA,
    const float* B, int ldb, long zsB, int trB, int K,
    const float* A2, int lda2, long zsA2,
    const float* B2, int ldb2, long zsB2, int trB2, int K2,
    const float* bias,
    const float* res, int ldres, long zsRes,
    const float* gmul, int ldg, int gdiv, float scale,
    float* D, int ldd, long zsD, int M, int N, int epi, int Z)
{
    const int strips = (N + 63) / 64;
    const int mtiles = (M + 15) / 16;
    dim3 grid((strips + 3) / 4, mtiles, Z);
    wmma_gemm<<<grid, dim3(128), 0, st>>>(
        A, lda, zsA, B, ldb, zsB, trB, K,
        A2, lda2, zsA2, B2, ldb2, zsB2, trB2, K2,
        bias, res, ldres, zsRes, gmul, ldg, gdiv, scale,
        D, ldd, zsD, M, N, epi);
}

extern "C" void kernel_launch(void* const* d_in, const int* in_sizes, int n_in,
                              void* d_out, int out_size, void* d_ws, size_t ws_size,
                              hipStream_t stream)
{
    const float* sc        = (const float*)d_in[0];   // (32,5,640)
    const float* bw        = (const float*)d_in[1];   // (32,351,640)
    const float* ss        = (const float*)d_in[2];   // (32,5,300)
    const float* bsem      = (const float*)d_in[3];   // (32,351,300)
    const float* bow       = (const float*)d_in[4];   // (32,351,640)
    const float* w_map1    = (const float*)d_in[5];
    const float* b_map1    = (const float*)d_in[6];
    const float* w_map2    = (const float*)d_in[7];
    const float* b_map2    = (const float*)d_in[8];
    const float* w_visfuse = (const float*)d_in[9];
    const float* b_visfuse = (const float*)d_in[10];
    const float* w_semfuse = (const float*)d_in[11];
    const float* b_semfuse = (const float*)d_in[12];
    const float* w_qs      = (const float*)d_in[13];
    const float* w_ks      = (const float*)d_in[14];
    const float* w_vs      = (const float*)d_in[15];
    const float* w_qs_sem  = (const float*)d_in[16];
    const float* w_ks_sem  = (const float*)d_in[17];
    const float* w_fc      = (const float*)d_in[18];
    const float* w_vs_open = (const float*)d_in[19];
    const float* w_fc_open = (const float*)d_in[20];

    float* ws = (float*)d_ws;
    float* ssem_h = ws;                      // 160*300
    float* ssem   = ssem_h + BROWS * SD;     // 160*300
    float* avg    = ssem   + BROWS * SD;     // 32*940
    float* gv     = avg    + BS * (FD + SD); // 32*640
    float* gs     = gv     + BS * FD;        // 32*300
    float* q      = gs     + BS * SD;        // 160*640
    float* qt     = q      + BROWS * FD;     // 160*640
    float* qtsem  = qt     + BROWS * FD;     // 160*300
    float* attn   = qtsem  + BROWS * SD;     // 160*352 (padded ld)
    float* ctx    = attn   + BROWS * LDATT;  // 160*640
    float* ctxo   = ctx    + BROWS * FD;     // 160*640
    float* t1     = ctxo   + BROWS * FD;     // 160*640
    float* outv   = t1     + BROWS * FD;     // 160*640
    float* t1o    = outv   + BROWS * FD;     // 160*640
    float* outo   = t1o    + BROWS * FD;     // 160*640

    const float invTemp = 1.0f / sqrtf((float)FD);

    // 1) per-batch mean over NBASE of [bw | bsem]  -> avg (32 x 940)
    avg_kernel<<<dim3(BS), dim3(256), 0, stream>>>(bw, bsem, avg);

    // 2) gates: sigmoid(avg @ W^T + b) + 1
    launch_gemm(stream, avg, FD + SD, 0, w_visfuse, FD + SD, 0, 0, FD + SD,
                nullptr, 0, 0, nullptr, 0, 0, 0, 0,
                b_visfuse, nullptr, 0, 0, nullptr, 0, 1, 1.0f,
                gv, FD, 0, BS, FD, EPI_SIGP1, 1);
    launch_gemm(stream, avg, FD + SD, 0, w_semfuse, FD + SD, 0, 0, FD + SD,
                nullptr, 0, 0, nullptr, 0, 0, 0, 0,
                b_semfuse, nullptr, 0, 0, nullptr, 0, 1, 1.0f,
                gs, SD, 0, BS, SD, EPI_SIGP1, 1);

    // 3) semantic calibration MLP: leaky_relu -> linear
    launch_gemm(stream, ss, SD, 0, w_map1, SD, 0, 0, SD,
                nullptr, 0, 0, nullptr, 0, 0, 0, 0,
                b_map1, nullptr, 0, 0, nullptr, 0, 1, 1.0f,
                ssem_h, SD, 0, BROWS, SD, EPI_LEAKY, 1);
    launch_gemm(stream, ssem_h, SD, 0, w_map2, SD, 0, 0, SD,
                nullptr, 0, 0, nullptr, 0, 0, 0, 0,
                b_map2, nullptr, 0, 0, nullptr, 0, 1, 1.0f,
                ssem, SD, 0, BROWS, SD, EPI_NONE, 1);

    // 4) q = sc @ w_qs^T + ssem @ w_qs_sem^T   (two fused pairs)
    launch_gemm(stream, sc, FD, 0, w_qs, FD, 0, 0, FD,
                ssem, SD, 0, w_qs_sem, SD, 0, 0, SD,
                nullptr, nullptr, 0, 0, nullptr, 0, 1, 1.0f,
                q, FD, 0, BROWS, FD, EPI_NONE, 1);

    // 5) fold key projection into query, apply gate and 1/TEMP:
    //    qt = (q @ w_ks) * gate_vis[batch] / TEMP ; qtsem = (q @ w_ks_sem) * gate_sem[batch] / TEMP
    launch_gemm(stream, q, FD, 0, w_ks, FD, 0, 1, FD,
                nullptr, 0, 0, nullptr, 0, 0, 0, 0,
                nullptr, nullptr, 0, 0, gv, FD, NWAY, invTemp,
                qt, FD, 0, BROWS, FD, EPI_GATE, 1);
    launch_gemm(stream, q, FD, 0, w_ks_sem, SD, 0, 1, FD,
                nullptr, 0, 0, nullptr, 0, 0, 0, 0,
                nullptr, nullptr, 0, 0, gs, SD, NWAY, invTemp,
                qtsem, SD, 0, BROWS, SD, EPI_GATE, 1);

    // 6) attention scores (batched over 32): score[b] = qt[b] @ bw[b]^T + qtsem[b] @ bsem[b]^T
    launch_gemm(stream,
                qt, FD, (long)NWAY * FD, bw, FD, (long)NBASE * FD, 0, FD,
                qtsem, SD, (long)NWAY * SD, bsem, SD, (long)NBASE * SD, 0, SD,
                nullptr, nullptr, 0, 0, nullptr, 0, 1, 1.0f,
                attn, LDATT, (long)NWAY * LDATT, NWAY, NBASE, EPI_NONE, BS);

    // 7) softmax over the 351 keys, per (b, way)
    softmax_kernel<<<dim3(BROWS), dim3(256), 0, stream>>>(attn);

    // 8) contract over NBASE first: ctx[b] = attn[b] @ bw[b], ctx_open[b] = attn[b] @ bow[b]
    launch_gemm(stream,
                attn, LDATT, (long)NWAY * LDATT, bw, FD, (long)NBASE * FD, 1, NBASE,
                nullptr, 0, 0, nullptr, 0, 0, 0, 0,
                nullptr, nullptr, 0, 0, nullptr, 0, 1, 1.0f,
                ctx, FD, (long)NWAY * FD, NWAY, FD, EPI_NONE, BS);
    launch_gemm(stream,
                attn, LDATT, (long)NWAY * LDATT, bow, FD, (long)NBASE * FD, 1, NBASE,
                nullptr, 0, 0, nullptr, 0, 0, 0, 0,
                nullptr, nullptr, 0, 0, nullptr, 0, 1, 1.0f,
                ctxo, FD, (long)NWAY * FD, NWAY, FD, EPI_NONE, BS);

    // 9) out = (ctx @ w_vs^T) @ w_fc^T + sc   (and the _open chain)
    launch_gemm(stream, ctx, FD, 0, w_vs, FD, 0, 0, FD,
                nullptr, 0, 0, nullptr, 0, 0, 0, 0,
                nullptr, nullptr, 0, 0, nullptr, 0, 1, 1.0f,
                t1, FD, 0, BROWS, FD, EPI_NONE, 1);
    launch_gemm(stream, t1, FD, 0, w_fc, FD, 0, 0, FD,
                nullptr, 0, 0, nullptr, 0, 0, 0, 0,
                nullptr, sc, FD, 0, nullptr, 0, 1, 1.0f,
                outv, FD, 0, BROWS, FD, EPI_RES, 1);
    launch_gemm(stream, ctxo, FD, 0, w_vs_open, FD, 0, 0, FD,
                nullptr, 0, 0, nullptr, 0, 0, 0, 0,
                nullptr, nullptr, 0, 0, nullptr, 0, 1, 1.0f,
                t1o, FD, 0, BROWS, FD, EPI_NONE, 1);
    launch_gemm(stream, t1o, FD, 0, w_fc_open, FD, 0, 0, FD,
                nullptr, 0, 0, nullptr, 0, 0, 0, 0,
                nullptr, sc, FD, 0, nullptr, 0, 1, 1.0f,
                outo, FD, 0, BROWS, FD, EPI_RES, 1);

    // 10) fake / fake_open means + passthrough of support_center
    finalize_kernel<<<dim3((BS * NWAY * FD + 255) / 256), dim3(256), 0, stream>>>(
        outv, outo, sc, (float*)d_out);

    (void)in_sizes; (void)n_in; (void)out_size; (void)ws_size;
}